// Model_55267639165014
// MI455X (gfx1250) — compile-verified
//
#include <hip/hip_runtime.h>
#include <math.h>

typedef float __attribute__((ext_vector_type(4))) v4f;
typedef int   __attribute__((ext_vector_type(4))) v4i;
typedef __attribute__((address_space(1))) v4i as1_v4i;
typedef __attribute__((address_space(3))) v4i as3_v4i;

// ---------- probe gfx1250 async global<->LDS builtins ----------
#if defined(__has_builtin)
#  if __has_builtin(__builtin_amdgcn_global_load_async_to_lds_b128) && \
      __has_builtin(__builtin_amdgcn_global_store_async_from_lds_b128)
#    define HAVE_ASYNC_LDS_COPY 1
#  endif
#  if __has_builtin(__builtin_amdgcn_s_wait_asynccnt)
#    define WAIT_ASYNC() __builtin_amdgcn_s_wait_asynccnt(0)
#  else
#    define WAIT_ASYNC() asm volatile("s_wait_asynccnt 0" ::: "memory")
#  endif
#endif

// ---------- monotone float <-> u32 key for atomic max on floats ----------
__device__ __forceinline__ unsigned f2key(float f) {
  unsigned u = __float_as_uint(f);
  return (u & 0x80000000u) ? ~u : (u | 0x80000000u);
}
__device__ __forceinline__ float key2f(unsigned k) {
  unsigned u = (k & 0x80000000u) ? (k & 0x7FFFFFFFu) : ~k;
  return __uint_as_float(u);
}

// ---------- ws init (deterministic each call) ----------
__global__ void init_ws_kernel(unsigned* ws) {
  if (threadIdx.x < 3) ws[threadIdx.x] = 0u;  // 0 < key(-inf): safe identity
}

// ---------- global max reduction (wave32 butterflies + atomic) ----------
__global__ __launch_bounds__(256)
void reduce_max_kernel(const float* __restrict__ x, long n, unsigned* slot) {
  float m = -INFINITY;
  long stride = (long)gridDim.x * blockDim.x;
  for (long i = (long)blockIdx.x * blockDim.x + threadIdx.x; i < n; i += stride)
    m = fmaxf(m, x[i]);
  #pragma unroll
  for (int off = 16; off; off >>= 1) m = fmaxf(m, __shfl_xor(m, off, 32));
  __shared__ float sm[8];
  int lane = threadIdx.x & 31, w = threadIdx.x >> 5;
  if (lane == 0) sm[w] = m;
  __syncthreads();
  if (w == 0) {
    m = (lane < (int)(blockDim.x >> 5)) ? sm[lane] : -INFINITY;
    #pragma unroll
    for (int off = 16; off; off >>= 1) m = fmaxf(m, __shfl_xor(m, off, 32));
    if (lane == 0) atomicMax(slot, f2key(m));
  }
}

// ---------- main kernel: one wave32 per batch row ----------
__global__ __launch_bounds__(256)
void splendor_kernel(const float* __restrict__ states,
                     const float* __restrict__ action_logits,
                     const float* __restrict__ discard_logits,
                     const float* __restrict__ noble_logits,
                     const float* __restrict__ card_cost,
                     const float* __restrict__ noble_info,
                     const float* __restrict__ take_info,
                     const unsigned* __restrict__ gmax_keys,
                     float* __restrict__ out_action_p,
                     float* __restrict__ out_discard_p,
                     float* __restrict__ out_noble_p,
                     float* __restrict__ out_action_idx,
                     float* __restrict__ out_discard_idx,
                     float* __restrict__ out_noble_idx,
                     int Bn) {
  __shared__ float s_cost[450];   // 90 x 5
  __shared__ float s_take[75];    // 15 x 5
  __shared__ float s_noble[50];   // 10 x 5
  for (int i = threadIdx.x; i < 450; i += blockDim.x) s_cost[i] = card_cost[i];
  for (int i = threadIdx.x; i < 75;  i += blockDim.x) s_take[i] = take_info[i];
  for (int i = threadIdx.x; i < 50;  i += blockDim.x) s_noble[i] = noble_info[i];
  __syncthreads();

  const float gmaxA = key2f(gmax_keys[0]);
  const float gmaxD = key2f(gmax_keys[1]);
  const float gmaxN = key2f(gmax_keys[2]);

  const int wave = threadIdx.x >> 5;
  const int lane = threadIdx.x & 31;
  const int row  = blockIdx.x * (blockDim.x >> 5) + wave;
  if (row >= Bn) return;

  const float* st = states + (long)row * 308;

  float cards_[5], res[5], csum[5];
  #pragma unroll
  for (int k = 0; k < 5; k++) {
    float ch = st[k];
    cards_[k] = st[6 + k];
    res[k]  = ch + cards_[k];
    csum[k] = ch + st[104 + k];        // VIEW_OTHER + CHIPS
  }
  const float gold = st[5];
  const float slot_ok = (st[101] + st[102] + st[103] < 3.0f) ? 1.0f : 0.0f;

  // ---- 199 actions, lane-strided (7 steps over wave32) ----
  const float* al = action_logits + (long)row * 199;
  float best = -INFINITY; int bidx = 0;
  #pragma unroll
  for (int k = 0; k < 7; k++) {
    int a = k * 32 + lane;
    if (a < 199) {
      float mask;
      if (a < 15) {                          // take actions
        bool ok = true;
        #pragma unroll
        for (int j = 0; j < 5; j++) ok = ok && (csum[j] + s_take[a * 5 + j] < 5.0f);
        mask = ok ? 1.0f : 0.0f;
        if (a >= 10) {                       // take-2 needs that color untouched
          int i2 = a - 10;
          float cz = (i2 == 0) ? csum[0] : (i2 == 1) ? csum[1] :
                     (i2 == 2) ? csum[2] : (i2 == 3) ? csum[3] : csum[4];
          mask *= (cz == 0.0f) ? 1.0f : 0.0f;
        }
      } else if (a == 15) {                  // pass
        mask = 1.0f;
      } else if (a < 106) {                  // purchase (board + reserved hand) * afford
        int c = a - 16;
        float sf = 0.0f;
        #pragma unroll
        for (int j = 0; j < 5; j++) sf += fmaxf(s_cost[c * 5 + j] - res[j], 0.0f);
        float aff = (sf <= gold) ? 1.0f : 0.0f;
        mask = (st[208 + c] + st[11 + c]) * aff;
      } else if (a < 196) {                  // reserve visible card
        mask = st[208 + (a - 106)] * slot_ok;
      } else {                               // reserve from deck top
        mask = slot_ok;
      }
      float p = expf(al[a] - gmaxA) * mask;
      out_action_p[(long)row * 199 + a] = p;
      if (p > best) { best = p; bidx = a; }  // ascending a => first-occurrence
    }
  }
  // cross-lane argmax (max value, min index among ties)
  #pragma unroll
  for (int off = 16; off; off >>= 1) {
    float ov = __shfl_xor(best, off, 32);
    int   oi = __shfl_xor(bidx, off, 32);
    if (ov > best || (ov == best && oi < bidx)) { best = ov; bidx = oi; }
  }
  if (lane == 0) out_action_idx[row] = (float)bidx;

  // ---- discard: 18 values (3 rows x 6 chips) ----
  float dv = 0.0f;
  if (lane < 18) {
    int j = lane % 6;
    float cm = (st[j] > 0.0f) ? 1.0f : 0.0f;
    dv = expf(discard_logits[(long)row * 18 + lane] - gmaxD) * cm;
    out_discard_p[(long)row * 18 + lane] = dv;
  }
  #pragma unroll
  for (int i = 0; i < 3; i++) {
    float b2 = -INFINITY; int bj = 0;
    #pragma unroll
    for (int j = 0; j < 6; j++) {
      float t = __shfl(dv, i * 6 + j, 32);   // all lanes execute (convergent)
      if (t > b2) { b2 = t; bj = j; }
    }
    if (lane == 0) out_discard_idx[(long)row * 3 + i] = (float)bj;
  }

  // ---- nobles: 10 values ----
  float nv = -INFINITY; int nidx = lane;
  if (lane < 10) {
    bool req = true;
    #pragma unroll
    for (int j = 0; j < 5; j++) req = req && (cards_[j] - s_noble[lane * 5 + j] >= 0.0f);
    float v = expf(noble_logits[(long)row * 10 + lane] - gmaxN) *
              st[298 + lane] * (req ? 1.0f : 0.0f);
    out_noble_p[(long)row * 10 + lane] = v;
    nv = v;
  }
  #pragma unroll
  for (int off = 16; off; off >>= 1) {
    float ov = __shfl_xor(nv, off, 32);
    int   oi = __shfl_xor(nidx, off, 32);
    if (ov > nv || (ov == nv && oi < nidx)) { nv = ov; nidx = oi; }
  }
  if (lane == 0) out_noble_idx[row] = (float)nidx;
}

// ---------- 128-bit non-temporal streaming copy (fallback path) ----------
__global__ __launch_bounds__(256)
void copy_mem_kernel(const v4f* __restrict__ src, v4f* __restrict__ dst, long n) {
  long stride = (long)gridDim.x * blockDim.x;
  for (long i = (long)blockIdx.x * blockDim.x + threadIdx.x; i < n; i += stride) {
    __builtin_prefetch(src + i + stride, 0, 0);       // global_prefetch_b8
    v4f v = __builtin_nontemporal_load(src + i);      // b128 load, NT hint
    __builtin_nontemporal_store(v, dst + i);          // b128 store, NT hint
  }
}

#if defined(HAVE_ASYNC_LDS_COPY)
// ---------- async global->LDS->global streaming copy (ASYNCcnt path) ----------
// Each block stages 4 * blockDim float4 tiles through LDS with
// global_load_async_to_lds_b128 / global_store_async_from_lds_b128.
// 4 ops in flight per wave; thousands of resident waves cover HBM latency.
__global__ __launch_bounds__(256)
void copy_mem_async_kernel(const float* __restrict__ src, float* __restrict__ dst,
                           long n4) {
  __shared__ v4f stage[4 * 256];   // 16 KB (of 320 KB/WGP)
  const int tid = threadIdx.x;
  const long tile = 4L * blockDim.x;               // float4s per block-iteration
  const long gstride = (long)gridDim.x * tile;
  for (long t0 = (long)blockIdx.x * tile; t0 < n4; t0 += gstride) {
    #pragma unroll
    for (int j = 0; j < 4; j++) {
      long idx = t0 + (long)j * blockDim.x + tid;  // contiguous addresses per op
      if (idx < n4) {
        __builtin_amdgcn_global_load_async_to_lds_b128(
            (as1_v4i*)(const_cast<float*>(src) + idx * 4),
            (as3_v4i*)&stage[j * 256 + tid], 0, 0);
      }
    }
    WAIT_ASYNC();                                  // loads landed in LDS
    #pragma unroll
    for (int j = 0; j < 4; j++) {
      long idx = t0 + (long)j * blockDim.x + tid;
      if (idx < n4) {
        __builtin_amdgcn_global_store_async_from_lds_b128(
            (as1_v4i*)(dst + idx * 4),
            (as3_v4i*)&stage[j * 256 + tid], 0, 0);
      }
    }
    WAIT_ASYNC();                                  // LDS reusable next iteration
  }
}
#endif

extern "C" void kernel_launch(void* const* d_in, const int* in_sizes, int n_in,
                              void* d_out, int out_size, void* d_ws, size_t ws_size,
                              hipStream_t stream) {
  const float* states = (const float*)d_in[0];
  const float* al     = (const float*)d_in[1];
  const float* dl     = (const float*)d_in[2];
  const float* nl     = (const float*)d_in[3];
  const float* mem    = (const float*)d_in[4];
  const float* cost   = (const float*)d_in[5];
  const float* ninfo  = (const float*)d_in[6];
  const float* tinfo  = (const float*)d_in[7];
  (void)n_in; (void)out_size; (void)ws_size;

  const long Bn = (long)(in_sizes[0] / 308);
  unsigned* ws = (unsigned*)d_ws;

  // output layout: action_p | discard_p | noble_p | action_idx | discard_idx | noble_idx | memory
  float* out   = (float*)d_out;
  float* o_ap  = out;
  float* o_dp  = o_ap + Bn * 199;
  float* o_np  = o_dp + Bn * 18;
  float* o_ai  = o_np + Bn * 10;
  float* o_di  = o_ai + Bn;
  float* o_ni  = o_di + Bn * 3;
  float* o_mem = o_ni + Bn;

  init_ws_kernel<<<1, 32, 0, stream>>>(ws);
  reduce_max_kernel<<<2048, 256, 0, stream>>>(al, Bn * 199, ws + 0);
  reduce_max_kernel<<<512,  256, 0, stream>>>(dl, Bn * 18,  ws + 1);
  reduce_max_kernel<<<256,  256, 0, stream>>>(nl, Bn * 10,  ws + 2);

  int rowsPerBlock = 8;  // 8 waves of 32 per block
  int grid = (int)((Bn + rowsPerBlock - 1) / rowsPerBlock);
  splendor_kernel<<<grid, 256, 0, stream>>>(states, al, dl, nl, cost, ninfo, tinfo,
                                            ws, o_ap, o_dp, o_np, o_ai, o_di, o_ni,
                                            (int)Bn);

  long n4 = Bn * 128;  // 512 floats/row -> 128 float4/row
#if defined(HAVE_ASYNC_LDS_COPY)
  copy_mem_async_kernel<<<2048, 256, 0, stream>>>(mem, o_mem, n4);
#else
  copy_mem_kernel<<<4096, 256, 0, stream>>>((const v4f*)mem, (v4f*)o_mem, n4);
#endif
}